// LocalityLoss_472446403064
// MI455X (gfx1250) — compile-verified
//
#include <hip/hip_runtime.h>
#include <hip/hip_bf16.h>
#include <cstdint>
#include <cstddef>

// ---------------------------------------------------------------------------
// LocalityLoss for feat_map (32, 512, 56, 56) f32.
// Phase 1: single streaming pass (205.5 MB -> ~8.8us at 23.3 TB/s), fused
//          accumulation of per-(t,h) and per-(t,w) {sum, sumsq} partials.
//          Tile row/col reductions of the accumulated planes run on the
//          matrix units via V_WMMA_F32_16X16X4_F32 with ones-vector operands.
//          Planes live in a zero-padded 64x68 LDS layout: unconditional
//          ds_loads (no EXEC diamonds) and conflict-free A-fragment banks.
// Phase 2: tiny finalize kernel (cumsums, sqrt, means) -> scalar.
// Fully deterministic: no atomics, fixed reduction order.
// Workspace use: 512 blocks * 224 floats = 458,752 bytes of d_ws.
// ---------------------------------------------------------------------------

typedef float v2f __attribute__((ext_vector_type(2)));
typedef float v4f __attribute__((ext_vector_type(4)));
typedef float v8f __attribute__((ext_vector_type(8)));

#define T_DIM 32
#define C_DIM 512
#define H_DIM 56
#define W_DIM 56
#define HW    (H_DIM * W_DIM)   /* 3136 */
#define NG    16                /* channel groups per batch element */
#define CG    (C_DIM / NG)      /* 32 channels per block */
#define NBLK  (T_DIM * NG)      /* 512 blocks */
#define PARTS 224               /* 4 arrays x 56 partials per block */
#define EPSF  1e-6f

#define LROWS   64              /* padded plane rows   */
#define LSTRIDE 68              /* padded plane stride in floats (4|68, 68%64=4) */
#define LPLANE  (LROWS * LSTRIDE)   /* 4352 floats = 17408 B per plane */
#define LP4     (LPLANE / 4)        /* 1088 float4 per plane */

// D = A(16x4 f32) * B(4x16 f32) + C(16x16 f32)
static __device__ __forceinline__ v8f wmma_f32(v2f a, v2f b, v8f c) {
  // 8 args: (neg_a, A, neg_b, B, c_mod, C, reuse_a, reuse_b)
  return __builtin_amdgcn_wmma_f32_16x16x4_f32(
      false, a, false, b, (short)0, c, false, false);
}

// Unconditional padded-plane reads (padding is zero-filled).
static __device__ __forceinline__ float ldp(const float* p, int r, int c) {
  return p[r * LSTRIDE + c];
}
static __device__ __forceinline__ v2f ldp2(const float* p, int r, int c) {
  return *(const v2f*)&p[r * LSTRIDE + c];   // (r*68+c)*4 is 8B aligned (c even)
}

__global__ __launch_bounds__(256)
void locality_partials(const float* __restrict__ feat, float* __restrict__ ws) {
  __shared__ float ldsL[LPLANE];          // per-block Sum_c x      (padded 64x68)
  __shared__ float ldsS[LPLANE];          // per-block Sum_c x*x    (padded 64x68)
  __shared__ float tRow[2][4][4][16];     // [plane L/S][ti][tj][m] row sums
  __shared__ float tCol[2][4][4][16];     // [plane L/S][ti][tj][n] col sums

  const int tid = threadIdx.x;
  const int t  = blockIdx.x / NG;
  const int g  = blockIdx.x % NG;

  // Zero both planes (covers the padding; data region is overwritten below).
  {
    const v4f z = {};
    for (int p = tid; p < LP4; p += 256) {
      ((v4f*)ldsS)[p] = z;
      ((v4f*)ldsL)[p] = z;
    }
  }
  __syncthreads();   // cheap: happens before the long streaming phase

  const float* base = feat + (size_t)(t * C_DIM + g * CG) * HW;

  // --- Phase A: stream CG channels, accumulate elementwise S/L in registers.
  // 784 float4 per plane; thread owns f = tid + 256*j (j=0..2) and, for
  // tid < 16, f = 768 + tid.
  v4f accS[4], accL[4];
#pragma unroll
  for (int j = 0; j < 4; ++j) { accS[j] = {}; accL[j] = {}; }

  for (int c = 0; c < CG; ++c) {
    const v4f* cb = (const v4f*)(base + (size_t)c * HW);
    if (c + 1 < CG) {
      // one-channel-ahead speculative prefetch (global_prefetch_b8)
      __builtin_prefetch(base + (size_t)(c + 1) * HW + tid * 4, 0, 1);
    }
#pragma unroll
    for (int j = 0; j < 4; ++j) {
      const int f = tid + 256 * j;
      if (j < 3 || tid < 16) {
        v4f x = __builtin_nontemporal_load(&cb[f]);  // stream-once data
        accS[j] += x * x;
        accL[j] += x;
      }
    }
  }

  // Scatter register accumulators into the padded planes. A row is 14 float4s,
  // so a float4 never straddles a row: f -> (h = f/14, w4 = f%14).
#pragma unroll
  for (int j = 0; j < 4; ++j) {
    const int f = tid + 256 * j;
    if (j < 3 || tid < 16) {
      const int h  = f / 14;
      const int w4 = f - h * 14;
      const int off4 = h * (LSTRIDE / 4) + w4;   // 16B-aligned (68 = 4*17)
      ((v4f*)ldsS)[off4] = accS[j];
      ((v4f*)ldsL)[off4] = accL[j];
    }
  }
  __syncthreads();

  // --- Phase B: WMMA row/col sums of the two padded planes.
  // 16 tiles of 16x16; wave w handles tiles 2w and 2w+1. EXEC is full here.
  const int wv   = tid >> 5;
  const int lane = tid & 31;
  const int mn   = lane & 15;          // row (A) / col (B) index within tile
  const int kb   = (lane >> 4) << 1;   // K sub-offset: lanes 16-31 hold K+2/K+3

#pragma unroll
  for (int q = 0; q < 2; ++q) {
    const int tile = wv * 2 + q;
    const int ti = tile >> 2, tj = tile & 3;
    const int h0 = ti * 16,  w0 = tj * 16;

    v8f rS = {}, rL = {}, cS = {}, cL = {};
    const v2f ones = {1.0f, 1.0f};

#pragma unroll
    for (int kc = 0; kc < 4; ++kc) {
      const int k0 = kc * 4 + kb;
      // A-fragment (ISA 16x4 f32 layout): lane<16 -> X[m][k0],X[m][k0+1];
      // lane>=16 -> X[m][k0+2],X[m][k0+3] (kb folds the +2). Contiguous pair
      // -> single ds_load_b64, conflict-free banks with stride 68.
      v2f aS = ldp2(ldsS, h0 + mn, w0 + k0);
      v2f aL = ldp2(ldsL, h0 + mn, w0 + k0);
      // B-fragment (4x16, mirrored layout): v0 = rows K/K+2, v1 = rows K+1/K+3.
      v2f bS, bL;
      bS.x = ldp(ldsS, h0 + k0,     w0 + mn);
      bS.y = ldp(ldsS, h0 + k0 + 1, w0 + mn);
      bL.x = ldp(ldsL, h0 + k0,     w0 + mn);
      bL.y = ldp(ldsL, h0 + k0 + 1, w0 + mn);

      rS = wmma_f32(aS, ones, rS);   // D[m][*] = sum_k S[m][k]  (row sums)
      cS = wmma_f32(ones, bS, cS);   // D[*][n] = sum_k S[k][n]  (col sums)
      rL = wmma_f32(aL, ones, rL);
      cL = wmma_f32(ones, bL, cL);
    }

    // Extract: D VGPR r holds (M=r, N=lane) for lanes 0-15 and (M=8+r,
    // N=lane-16) for lanes 16-31. Row sums are constant across N -> take N=0.
    if ((lane & 15) == 0) {
      const int mb = (lane >> 4) * 8;
#pragma unroll
      for (int r = 0; r < 8; ++r) {
        tRow[0][ti][tj][mb + r] = rL[r];
        tRow[1][ti][tj][mb + r] = rS[r];
      }
    }
    // Col sums are constant across M -> VGPR0, lanes 0-15 (N = lane).
    if (lane < 16) {
      tCol[0][ti][tj][lane] = cL[0];
      tCol[1][ti][tj][lane] = cS[0];
    }
  }
  __syncthreads();

  // --- Phase C: fixed-order combine across tiles, write 224 partials.
  // arr: 0=lin_h 1=sq_h 2=lin_w 3=sq_w ; plane 0=L(lin) 1=S(sq).
  if (tid < PARTS) {
    const int arr = tid / 56;
    const int idx = tid - arr * 56;
    const int plane = arr & 1;
    const int tt = idx >> 4, sub = idx & 15;
    float s = 0.0f;
    if (arr < 2) {
#pragma unroll
      for (int j = 0; j < 4; ++j) s += tRow[plane][tt][j][sub];
    } else {
#pragma unroll
      for (int i = 0; i < 4; ++i) s += tCol[plane][i][tt][sub];
    }
    ws[(size_t)blockIdx.x * PARTS + tid] = s;
  }
}

__global__ __launch_bounds__(32)
void locality_finalize(const float* __restrict__ ws, float* __restrict__ out) {
  __shared__ float bufS[32][56];
  __shared__ float bufL[32][56];
  __shared__ float red[32];

  const int t = threadIdx.x;            // one batch element per lane
  const float nOther = (float)(C_DIM * 56);   // 512*56 = 28672
  float acc = 0.0f;

  for (int axis = 0; axis < 2; ++axis) {
    const int arrL = axis * 2;          // lin array index (0 or 2)
    const int arrS = axis * 2 + 1;      // sq  array index (1 or 3)

    // Aggregate the 16 channel-group partials (fixed order).
    for (int i = 0; i < 56; ++i) {
      const float* pS = ws + (size_t)t * NG * PARTS + arrS * 56 + i;
      const float* pL = ws + (size_t)t * NG * PARTS + arrL * 56 + i;
      float sS = 0.0f, sL = 0.0f;
#pragma unroll
      for (int gg = 0; gg < NG; ++gg) {
        sS += pS[gg * PARTS];
        sL += pL[gg * PARTS];
      }
      bufS[t][i] = sS;
      bufL[t][i] = sL;
    }
    // Prefix pass: positions [0..i]
    float cs = 0.0f, cl = 0.0f;
    for (int i = 0; i < 56; ++i) {
      cs += bufS[t][i];
      cl += bufL[t][i];
      const float n = nOther * (float)(i + 1);
      acc += sqrtf(cs + 2.0f * EPSF * cl + (EPSF * EPSF) * n) + EPSF;
    }
    // Suffix pass: positions [i..55]
    cs = 0.0f; cl = 0.0f;
    for (int i = 55; i >= 0; --i) {
      cs += bufS[t][i];
      cl += bufL[t][i];
      const float n = nOther * (float)(56 - i);
      acc += sqrtf(cs + 2.0f * EPSF * cl + (EPSF * EPSF) * n) + EPSF;
    }
  }

  red[t] = acc;
  __syncthreads();
  if (t == 0) {
    float s = 0.0f;
    for (int i = 0; i < 32; ++i) s += red[i];   // fixed order -> deterministic
    // (L1+L2+L3+L4)/4 = (1/(32*4)) * sum_t acc_t
    out[0] = s * (1.0f / 128.0f);
  }
}

extern "C" void kernel_launch(void* const* d_in, const int* in_sizes, int n_in,
                              void* d_out, int out_size, void* d_ws, size_t ws_size,
                              hipStream_t stream) {
  (void)in_sizes; (void)n_in; (void)out_size; (void)ws_size;
  const float* feat = (const float*)d_in[0];
  float* out = (float*)d_out;
  float* ws  = (float*)d_ws;   // needs 512*224*4 = 458,752 bytes

  locality_partials<<<dim3(NBLK), dim3(256), 0, stream>>>(feat, ws);
  locality_finalize<<<dim3(1), dim3(32), 0, stream>>>(ws, out);
}